// RESCAL_78211354460368
// MI455X (gfx1250) — compile-verified
//
#include <hip/hip_runtime.h>
#include <math.h>

typedef __attribute__((ext_vector_type(2))) float v2f;
typedef __attribute__((ext_vector_type(8))) float v8f;

#define N_NODES     50000
#define N_EDGES     100000
#define N_CHAN      64
#define N_REL       16
#define EDGES_PER_BLOCK 64            // 4 sub-tiles of 16 edges share one R staging
#define MAX_TILES   (N_EDGES / EDGES_PER_BLOCK + N_REL)   // 1578 upper bound

// ws layout (ints): [0..15] counts, [16..32] offsets(17), [33..49] tileOff(17),
//                   [50..65] cursors, [128..128+N_EDGES) perm
#define WS_COUNTS   0
#define WS_OFFSETS  16
#define WS_TILEOFF  33
#define WS_CURSORS  50
#define WS_PERM     128

// ---------------- bucketing kernels ----------------

__global__ void k_zero(int* w) {
    int t = threadIdx.x;
    if (t < N_REL) w[WS_COUNTS + t] = 0;
}

__global__ void k_count(const int* __restrict__ et, int* w) {
    int e = blockIdx.x * blockDim.x + threadIdx.x;
    if (e < N_EDGES) atomicAdd(&w[WS_COUNTS + et[e]], 1);
}

__global__ void k_scan(int* w) {
    if (threadIdx.x != 0) return;
    int off = 0, tiles = 0;
    w[WS_OFFSETS + 0] = 0;
    w[WS_TILEOFF + 0] = 0;
    for (int r = 0; r < N_REL; ++r) {
        int c = w[WS_COUNTS + r];
        w[WS_CURSORS + r] = off;
        off += c;
        tiles += (c + EDGES_PER_BLOCK - 1) / EDGES_PER_BLOCK;
        w[WS_OFFSETS + r + 1] = off;
        w[WS_TILEOFF + r + 1] = tiles;
    }
}

__global__ void k_scatter(const int* __restrict__ et, int* w) {
    int e = blockIdx.x * blockDim.x + threadIdx.x;
    if (e < N_EDGES) {
        int r = et[e];
        int pos = atomicAdd(&w[WS_CURSORS + r], 1);
        w[WS_PERM + pos] = e;
    }
}

// ---------------- WMMA compute kernel ----------------
// One wave32 per 64-edge super-tile (all edges share relation r).
// R is staged to LDS ONCE, then 4 sub-tiles of 16 edges run:
//   T = R * O   (64x16) : 4 row-blocks x 16 k-steps of v_wmma_f32_16x16x4_f32
//   D = S^T * T (16x16) : 16 k-steps; diag(D) = scores
// LDS row strides padded to 68 floats => lanes hit distinct banks (bank step 4).

#define RS 68   // padded row stride (68*4=272B, 16B aligned, conflict-free reads)

__global__ __launch_bounds__(32) void k_rescal_wmma(
    const float* __restrict__ x,
    const int*   __restrict__ ei,     // [2][N_EDGES]
    const float* __restrict__ W,      // [16][64][64]
    const int*   __restrict__ w,      // workspace ints
    float*       __restrict__ out)
{
    __shared__ float r_lds[64 * RS];      // R matrix, padded (staged once per block)
    __shared__ float s_lds[16 * RS];      // normalized subject vectors
    __shared__ float o_lds[16 * RS];      // normalized object vectors
    __shared__ float t_lds[64 * 16];      // T = R*O
    __shared__ float d_lds[16 * 17];      // D = S^T*T (padded)

    const int* tileOff = w + WS_TILEOFF;
    const int* offsets = w + WS_OFFSETS;
    const int* perm    = w + WS_PERM;

    int b = blockIdx.x;
    if (b >= tileOff[N_REL]) return;      // uniform exit

    // locate relation bucket (uniform scalar loop over 16 entries)
    int r = 0;
    while (r < N_REL - 1 && b >= tileOff[r + 1]) ++r;
    int t    = b - tileOff[r];
    int base = offsets[r] + t * EDGES_PER_BLOCK;
    int cnt  = offsets[r + 1] - offsets[r];
    int rem  = cnt - t * EDGES_PER_BLOCK;             // 1..64 edges in this super-tile

    int lane = threadIdx.x;
    int half = lane >> 4;     // 0: K={0,1} / subject side, 1: K={2,3} / object side
    int lm   = lane & 15;

    // ---- stage R into LDS once (1024 float4 loads across 32 lanes) ----
    const float* Wr = W + r * (N_CHAN * N_CHAN);
    for (int i = lane; i < 1024; i += 32) {
        int row = i >> 4;
        int c4  = (i & 15) << 2;
        float4 v = *(const float4*)(Wr + row * N_CHAN + c4);
        *(float4*)(&r_lds[row * RS + c4]) = v;
    }
    __syncthreads();

    // ---- loop over 4 sub-tiles of 16 edges sharing this R ----
    for (int sub = 0; sub < EDGES_PER_BLOCK / 16 && sub * 16 < rem; ++sub) {
        int nvalid = rem - sub * 16; if (nvalid > 16) nvalid = 16;
        int estart = base + sub * 16;

        // gather + normalize: lanes 0-15 -> s of edge lm, lanes 16-31 -> o
        int eid = 0;
        {
            float* dst = half ? &o_lds[lm * RS] : &s_lds[lm * RS];
            if (lm < nvalid) {
                eid = perm[estart + lm];
                int node = half ? ei[N_EDGES + eid] : ei[eid];
                const float* row = x + (long)node * N_CHAN;
                float ss = 0.0f;
                for (int c = 0; c < N_CHAN; c += 4) {
                    float4 v = *(const float4*)(row + c);
                    *(float4*)(dst + c) = v;
                    ss += v.x * v.x + v.y * v.y + v.z * v.z + v.w * v.w;
                }
                float rinv = ss > 0.0f ? 1.0f / sqrtf(ss) : 0.0f;
                for (int c = 0; c < N_CHAN; c += 4) {
                    float4 v = *(float4*)(dst + c);
                    v.x *= rinv; v.y *= rinv; v.z *= rinv; v.w *= rinv;
                    *(float4*)(dst + c) = v;
                }
            } else {
                float4 z = {0.f, 0.f, 0.f, 0.f};
                for (int c = 0; c < N_CHAN; c += 4) *(float4*)(dst + c) = z;
            }
        }
        __syncthreads();

        // stage 1: T(64x16) = R(64x64) * O(64x16), fp32 WMMA, K=4 steps
        // A frag (16x4): lane lm = row, VGPR0/1 = K (half*2 +0/+1)
        // B frag (4x16): lane lm = col, VGPR0/1 = K (half*2 +0/+1)
        for (int rb = 0; rb < 4; ++rb) {
            v8f acc = {0.f, 0.f, 0.f, 0.f, 0.f, 0.f, 0.f, 0.f};
            const float* ra = &r_lds[(rb * 16 + lm) * RS];
            #pragma unroll
            for (int kk = 0; kk < 16; ++kk) {
                int k = kk * 4 + half * 2;
                v2f a, bb;
                a.x  = ra[k];
                a.y  = ra[k + 1];
                bb.x = o_lds[lm * RS + k];
                bb.y = o_lds[lm * RS + k + 1];
                acc = __builtin_amdgcn_wmma_f32_16x16x4_f32(
                    false, a, false, bb, (short)0, acc, false, false);
            }
            // D layout: VGPR j holds row (half*8 + j), col lm
            #pragma unroll
            for (int j = 0; j < 8; ++j)
                t_lds[(rb * 16 + half * 8 + j) * 16 + lm] = acc[j];
        }
        __syncthreads();

        // stage 2: D(16x16) = S^T(16x64) * T(64x16)
        v8f d = {0.f, 0.f, 0.f, 0.f, 0.f, 0.f, 0.f, 0.f};
        #pragma unroll
        for (int kk = 0; kk < 16; ++kk) {
            int k = kk * 4 + half * 2;
            v2f a, bb;
            a.x  = s_lds[lm * RS + k];
            a.y  = s_lds[lm * RS + k + 1];
            bb.x = t_lds[k * 16 + lm];
            bb.y = t_lds[(k + 1) * 16 + lm];
            d = __builtin_amdgcn_wmma_f32_16x16x4_f32(
                false, a, false, bb, (short)0, d, false, false);
        }
        #pragma unroll
        for (int j = 0; j < 8; ++j)
            d_lds[(half * 8 + j) * 17 + lm] = d[j];
        __syncthreads();

        // diag(D)[e] = score of edge e in this sub-tile
        if (half == 0 && lm < nvalid)
            out[eid] = d_lds[lm * 17 + lm];
        __syncthreads();
    }
}

// ---------------- launcher ----------------

extern "C" void kernel_launch(void* const* d_in, const int* in_sizes, int n_in,
                              void* d_out, int out_size, void* d_ws, size_t ws_size,
                              hipStream_t stream) {
    const float* x  = (const float*)d_in[0];
    const int*   ei = (const int*)d_in[1];   // [2][N_EDGES]
    const int*   et = (const int*)d_in[2];   // [N_EDGES]
    const float* W  = (const float*)d_in[3]; // [16][64][64]
    float* out = (float*)d_out;
    int*   w   = (int*)d_ws;

    int nb = (N_EDGES + 255) / 256;
    k_zero   <<<1, 64, 0, stream>>>(w);
    k_count  <<<nb, 256, 0, stream>>>(et, w);
    k_scan   <<<1, 1, 0, stream>>>(w);
    k_scatter<<<nb, 256, 0, stream>>>(et, w);
    k_rescal_wmma<<<MAX_TILES, 32, 0, stream>>>(x, ei, W, w, out);
}